// bayesian_categorical_crossentropy_9474697855442
// MI455X (gfx1250) — compile-verified
//
#include <hip/hip_runtime.h>
#include <hip/hip_bf16.h>
#include <math.h>

// Problem constants (fixed instance: T=30, 2 classes, N from in_sizes).
constexpr int   T_SAMPLES = 30;
constexpr float F_EPS     = 1e-15f;

typedef __attribute__((ext_vector_type(2))) float v2f;
typedef __attribute__((ext_vector_type(8))) float v8f;

// Full-precision f32 wave32 sum using V_WMMA_F32_16X16X4_F32.
// A: a0 = per-lane value, a1 = 0  ->  A[m][0]=v[m], A[m][2]=v[m+16], rest 0.
// B: all ones -> D[m][n] = rowsum_m = v[m] + v[m+16] (layout-independent in B).
// D layout (documented): VGPR i = row (i + 8*(lane/16)).
// Sum of the 8 D regs per lane = half-wave sum; shfl_xor(16) completes it.
__device__ __forceinline__ float wave_reduce_sum_wmma(float v) {
  v2f a; a.x = v;    a.y = 0.0f;
  v2f b; b.x = 1.0f; b.y = 1.0f;
  v8f c = {};
  c = __builtin_amdgcn_wmma_f32_16x16x4_f32(
      /*neg_a=*/false, a, /*neg_b=*/false, b,
      /*c_mod=*/(short)0, c, /*reuse_a=*/false, /*reuse_b=*/false);
  float h = c.s0 + c.s1 + c.s2 + c.s3 + c.s4 + c.s5 + c.s6 + c.s7;
  h += __shfl_xor(h, 16);   // combine the two half-wave row groups
  return h;                 // total present in every lane
}

// 2-class cross-entropy picked-logprob term: lse(x0,x1) - x[label]
__device__ __forceinline__ float ce2(float x0, float x1, int y) {
  float m   = fmaxf(x0, x1);
  float dmn = fminf(x0, x1) - m;            // <= 0
  float lse = m + log1pf(expf(dmn));
  return lse - (y ? x1 : x0);
}

// Kernel 1: grid-stride streaming pass. 32 f32 accumulators per thread:
//   acc[0..29] = sum_n ce over distorted sample t
//   acc[30]    = sum_n undistorted ce
//   acc[31]    = sum_n expm1(var)
// Block (256 thr = 8 waves) reduces via per-wave WMMA f32 reduction + LDS,
// writes 32 partials per block into d_ws. Fully deterministic tree.
__global__ __launch_bounds__(256) void bcc_main_kernel(
    const float* __restrict__ logit, const float* __restrict__ var,
    const int* __restrict__ label, const float* __restrict__ noise,
    float* __restrict__ partial, int N) {
  float acc[32];
#pragma unroll
  for (int c = 0; c < 32; ++c) acc[c] = 0.0f;

  const float2* l2 = reinterpret_cast<const float2*>(logit);
  const float2* nz = reinterpret_cast<const float2*>(noise);

  const int stride = gridDim.x * blockDim.x;
  for (int n = blockIdx.x * blockDim.x + threadIdx.x; n < N; n += stride) {
    float2 l = l2[n];                 // global_load_b64, coalesced
    float  v = var[n];
    int    y = label[n];
    float sd = sqrtf(v) + F_EPS;
    float x0 = l.x + F_EPS;
    float x1 = l.y + F_EPS;

    acc[30] += ce2(x0, x1, y);
    acc[31] += expm1f(v);

#pragma unroll
    for (int t = 0; t < T_SAMPLES; ++t) {
      // wave reads 256 contiguous bytes per t: fully coalesced b64 stream
      float2 e = nz[(size_t)t * (size_t)N + (size_t)n];
      acc[t] += ce2(x0 + sd * e.x, x1 + sd * e.y, y);
    }
  }

  __shared__ float red[8][32];
  const int lane = threadIdx.x & 31;
  const int wid  = threadIdx.x >> 5;

#pragma unroll
  for (int c = 0; c < 32; ++c) {
    float tot = wave_reduce_sum_wmma(acc[c]);   // EXEC all-1s here
    if (lane == 0) red[wid][c] = tot;
  }
  __syncthreads();

  if (threadIdx.x < 32) {
    float s = 0.0f;
#pragma unroll
    for (int w = 0; w < 8; ++w) s += red[w][threadIdx.x];
    partial[(size_t)blockIdx.x * 32 + threadIdx.x] = s;
  }
}

// Kernel 2: single block reduces [nb x 32] partials, then does the tiny
// T-length ELU combine and writes the scalar.
__global__ __launch_bounds__(256) void bcc_final_kernel(
    const float* __restrict__ partial, float* __restrict__ out,
    int nb, float invN) {
  __shared__ float red[8][32];
  __shared__ float tot[32];
  const int lane = threadIdx.x & 31;
  const int wid  = threadIdx.x >> 5;

  float s = 0.0f;
  for (int r = wid; r < nb; r += 8) s += partial[(size_t)r * 32 + lane];
  red[wid][lane] = s;
  __syncthreads();

  if (threadIdx.x < 32) {
    float t = 0.0f;
#pragma unroll
    for (int w = 0; w < 8; ++w) t += red[w][threadIdx.x];
    tot[threadIdx.x] = t;
  }
  __syncthreads();

  if (threadIdx.x == 0) {
    float u   = tot[30] * invN;   // undistorted_loss
    float vdm = tot[31] * invN;   // mean variance_depressor
    float mc_sum = 0.0f;
#pragma unroll
    for (int t = 0; t < T_SAMPLES; ++t) {
      float d = tot[t] * invN;    // distorted_loss[t]
      float x = u - d;
      float elu = (x > 0.0f) ? x : expm1f(x);
      mc_sum += -elu;             // monte_carlo[t]
    }
    float variance_loss = (mc_sum / (float)T_SAMPLES) * u;
    out[0] = variance_loss + u + vdm;
  }
}

extern "C" void kernel_launch(void* const* d_in, const int* in_sizes, int n_in,
                              void* d_out, int out_size, void* d_ws, size_t ws_size,
                              hipStream_t stream) {
  const float* logit = (const float*)d_in[0];   // [N, 2] f32
  const float* var   = (const float*)d_in[1];   // [N, 1] f32
  const int*   label = (const int*)d_in[2];     // [N]    i32
  const float* noise = (const float*)d_in[3];   // [T, N, 2] f32

  const int N = in_sizes[0] / 2;

  int nb = 1024;                                // 262144 threads, 4 n per thread
  size_t cap = ws_size / (32 * sizeof(float));
  if ((size_t)nb > cap) nb = (int)cap;
  if (nb < 1) nb = 1;

  float* partial = (float*)d_ws;                // [nb, 32] f32

  bcc_main_kernel<<<nb, 256, 0, stream>>>(logit, var, label, noise, partial, N);
  bcc_final_kernel<<<1, 256, 0, stream>>>(partial, (float*)d_out, nb,
                                          1.0f / (float)N);
}